// GMMBlock_27315992002965
// MI455X (gfx1250) — compile-verified
//
#include <hip/hip_runtime.h>

// GMM per-pixel update, B=8 C=3 K=4 H=512 W=512, fp32 in/out.
// Pure HBM-streaming kernel: 654 MB traffic -> ~28us floor @ 23.3 TB/s.
// float4 (b128) coalesced NT loads/stores; K=4 fully unrolled in registers.
// Trans-pipe minimized: one v_log per (k,j) via log(n*rsqrt(s2)) fusion.

typedef __attribute__((ext_vector_type(4))) float v4f;

#define GB 8
#define GC 3
#define GK 4
#define GH 512
#define GW 512

__global__ __launch_bounds__(256) void gmm_update_kernel(
    const float* __restrict__ x,
    const float* __restrict__ U,
    const float* __restrict__ V,
    const float* __restrict__ N,
    const float* __restrict__ etap,
    float* __restrict__ out)
{
    constexpr int HW4  = (GH * GW) / 4;          // 65536 float4 per (b, ck) plane
    constexpr int NU4  = GB * GC * GK * HW4;     // float4 count of U/V/N (6,291,456)

    const int t   = blockIdx.x * blockDim.x + threadIdx.x;  // one float4 site
    const int hw4 = t & (HW4 - 1);
    const int bc  = t >> 16;                     // t / HW4
    const int base = bc * (GK * HW4) + hw4;      // float4 offset of k=0 plane

    const v4f* x4 = (const v4f*)x;
    const v4f* U4 = (const v4f*)U;
    const v4f* V4 = (const v4f*)V;
    const v4f* N4 = (const v4f*)N;
    v4f* o4 = (v4f*)out;                         // [U_out | V_out | N_out | prob]

    // Streaming loads, non-temporal: working set (654MB) >> 192MB L2.
    v4f xv = __builtin_nontemporal_load(x4 + t);
    v4f Uv[GK], Vv[GK], Nv[GK];
#pragma unroll
    for (int k = 0; k < GK; ++k) {
        const int off = base + k * HW4;
        Uv[k] = __builtin_nontemporal_load(U4 + off);
        Vv[k] = __builtin_nontemporal_load(V4 + off);
        Nv[k] = __builtin_nontemporal_load(N4 + off);
    }

    const float eta  = etap[0];
    const float nTot = 1.0f / eta - 1.0f;

    v4f Uo[GK], Vo[GK], No[GK], probv;

#pragma unroll
    for (int j = 0; j < 4; ++j) {
        const float xs  = xv[j];
        const float xs2 = xs * xs;

        float zz[GK], lp[GK], nn[GK];
        float sumN = 0.0f;
#pragma unroll
        for (int k = 0; k < GK; ++k) {
            const float u  = Uv[k][j];
            const float v  = Vv[k][j];
            const float n  = Nv[k][j];
            const float s2 = fmaxf(v - u * u, 0.01f);
            const float rs = __builtin_amdgcn_rsqf(s2);      // 1/S
            const float z  = (xs - u) * rs;
            zz[k] = z;
            // log_prob up to a k-independent constant (cancels in softmax):
            //   log(Nn_k) - 0.5 z^2 - log(S)
            // = log(n_k) + const - 0.5 z^2 + log(1/S)
            // = log(n_k * rs) - 0.5 z^2 + const            (single v_log)
            lp[k] = __logf(n * rs) - 0.5f * z * z;
            nn[k] = n;
            sumN += n;
        }
        const float rsum = __builtin_amdgcn_rcpf(sumN);

        // prob = (1/sumN) * sum_k n_k * Phi(|z_k|)
        float pr = 0.0f;
#pragma unroll
        for (int k = 0; k < GK; ++k) {
            const float az  = fabsf(zz[k]);
            const float cdf = 0.5f * (1.0f + erff(az * 0.70710678118654752f));
            pr += nn[k] * cdf;
        }
        probv[j] = pr * rsum;

        // softmax over K=4 (max-subtracted for stability)
        const float m = fmaxf(fmaxf(lp[0], lp[1]), fmaxf(lp[2], lp[3]));
        float e[GK], den = 0.0f;
#pragma unroll
        for (int k = 0; k < GK; ++k) { e[k] = __expf(lp[k] - m); den += e[k]; }
        const float rden = __builtin_amdgcn_rcpf(den);

        const float nScale = nTot * rsum;   // Nn_k = nScale * n_k
#pragma unroll
        for (int k = 0; k < GK; ++k) {
            const float Gam  = e[k] * rden;
            const float Nout = nScale * nn[k] + Gam;
            const float Et   = Gam * __builtin_amdgcn_rcpf(Nout);
            const float u    = Uv[k][j];
            const float v    = Vv[k][j];
            Uo[k][j] = u + Et * (xs - u);
            Vo[k][j] = v + Et * (xs2 - v);
            No[k][j] = Nout;
        }
    }

    // Streaming stores, non-temporal.
#pragma unroll
    for (int k = 0; k < GK; ++k) {
        const int off = base + k * HW4;
        __builtin_nontemporal_store(Uo[k], o4 + off);
        __builtin_nontemporal_store(Vo[k], o4 + NU4 + off);
        __builtin_nontemporal_store(No[k], o4 + 2 * NU4 + off);
    }
    __builtin_nontemporal_store(probv, o4 + 3 * NU4 + t);
}

extern "C" void kernel_launch(void* const* d_in, const int* in_sizes, int n_in,
                              void* d_out, int out_size, void* d_ws, size_t ws_size,
                              hipStream_t stream) {
    (void)in_sizes; (void)n_in; (void)out_size; (void)d_ws; (void)ws_size;
    const float* x   = (const float*)d_in[0];
    const float* U   = (const float*)d_in[1];
    const float* V   = (const float*)d_in[2];
    const float* N   = (const float*)d_in[3];
    const float* eta = (const float*)d_in[4];
    float* out = (float*)d_out;

    constexpr int total4 = GB * GC * ((GH * GW) / 4);  // 1,572,864 float4 sites
    constexpr int block  = 256;                        // 8 wave32 waves
    constexpr int grid   = total4 / block;             // 6144, exact

    gmm_update_kernel<<<grid, block, 0, stream>>>(x, U, V, N, eta, out);
}